// Beam_70128226009498
// MI455X (gfx1250) — compile-verified
//
#include <hip/hip_runtime.h>
#include <math.h>
#include <stdint.h>

typedef __attribute__((ext_vector_type(2))) float v2f;
typedef __attribute__((ext_vector_type(8))) float v8f;

#define B_SZ 16
#define N_SZ 16
#define V_SZ 32000
#define D_SZ 512
#define K_BEAM 4
#define ROWS 64                 // B*K
#define END_TOK 1
#define BIG_NEG (-1.0e9f)
#define NEG_INF (-3.0e38f)
#define VTILE 128
#define T_TILES (V_SZ / VTILE)  // 250
#define KCHUNK 128

// workspace byte offsets (total ~3.2 MB)
#define OFF_CL     0u
#define OFF_PMAX   2048000u
#define OFF_PSUM   2112000u
#define OFF_PTOPV  2176000u
#define OFF_PTOPI  2688000u
#define OFF_TGT    3200000u
#define OFF_OUT    3204096u
#define OFF_LOGP   3208192u
#define OFF_SCORE  3208448u
#define OFF_FLAG   3208704u
#define OFF_CONT   3208960u

__device__ __forceinline__ void ins8(float v, int idx, float tv[8], int ti[8]) {
  float cv = v; int ci = idx;
  #pragma unroll
  for (int q = 0; q < 8; ++q) {
    if (cv > tv[q]) {
      float ov = tv[q]; int oi = ti[q];
      tv[q] = cv; ti[q] = ci;
      cv = ov; ci = oi;
    }
  }
}

__device__ __forceinline__ void top4sel(const float vals[8], int sel[4]) {
  bool used[8];
  #pragma unroll
  for (int q = 0; q < 8; ++q) used[q] = false;
  #pragma unroll
  for (int j = 0; j < 4; ++j) {
    float best = -3.9e38f; int bj = 0;
    #pragma unroll
    for (int q = 0; q < 8; ++q)
      if (!used[q] && vals[q] > best) { best = vals[q]; bj = q; }
    used[bj] = true; sel[j] = bj;
  }
}

// async memory->LDS copy of 16 bytes per lane (GLOBAL_LOAD_ASYNC_TO_LDS_B128,
// tracked by ASYNCcnt).  ldsOff = addrspace(3) byte offset, gaddr = global addr.
__device__ __forceinline__ void async_copy_b128(unsigned ldsOff, const void* gaddr) {
  asm volatile("global_load_async_to_lds_b128 %0, %1, off"
               :
               : "v"(ldsOff), "v"((unsigned long long)(uintptr_t)gaddr)
               : "memory");
}

__device__ __forceinline__ void wait_async_zero() {
  asm volatile("s_wait_asynccnt 0x0" ::: "memory");
}

// ---------------- init beam-search state ----------------
__global__ void init_state_kernel(const int* __restrict__ x,
                                  int* __restrict__ tgt, int* __restrict__ out,
                                  float* __restrict__ logp, float* __restrict__ score,
                                  int* __restrict__ flag, int* __restrict__ cont) {
  const int tid = threadIdx.x;
  for (int g = tid; g < ROWS * N_SZ; g += blockDim.x) {
    int r = g / N_SZ, n = g % N_SZ;
    int v = x[(r >> 2) * N_SZ + n];
    tgt[g] = v; out[g] = v;
  }
  if (tid < ROWS) {
    logp[tid]  = ((tid & 3) == 0) ? 0.0f : BIG_NEG;
    score[tid] = BIG_NEG;
    flag[tid]  = 0;
  }
  if (tid == 0) *cont = 1;
}

// ---------------- CL = ctx @ Wctx  (16 x 32000, WMMA f32 16x16x4) ----------------
__global__ void ctx_logits_kernel(const float* __restrict__ ctx,
                                  const float* __restrict__ Wctx,
                                  float* __restrict__ CL) {
  const int tid  = threadIdx.x;
  const int w    = tid >> 5;
  const int lane = tid & 31;
  const int lrow = lane & 15;
  const int lhi  = lane >> 4;
  const int v    = blockIdx.x * 64 + w * 16 + lrow;
  const float* ap = ctx + lrow * D_SZ + 2 * lhi;
  v8f acc = {0.f,0.f,0.f,0.f,0.f,0.f,0.f,0.f};
  for (int kk = 0; kk < D_SZ; kk += 4) {
    v2f a; a.x = ap[kk]; a.y = ap[kk + 1];
    const float* bp = Wctx + (size_t)(kk + 2 * lhi) * V_SZ + v;
    v2f b; b.x = bp[0]; b.y = bp[V_SZ];
    acc = __builtin_amdgcn_wmma_f32_16x16x4_f32(false, a, false, b, (short)0, acc,
                                                false, false);
  }
  #pragma unroll
  for (int j = 0; j < 8; ++j)
    CL[(size_t)(j + 8 * lhi) * V_SZ + v] = acc[j];
}

// ------- per-step: z = emb[prev_tok] @ Wout + CL ; per-(row,tile) stats + top8 -------
__global__ void logits_topk_kernel(const float* __restrict__ emb,
                                   const float* __restrict__ Wout,
                                   const float* __restrict__ CL,
                                   const int* __restrict__ tgt, int step,
                                   float* __restrict__ pmax, float* __restrict__ psum,
                                   float* __restrict__ ptopv, int* __restrict__ ptopi) {
  extern __shared__ float lds[];
  float* ldsA = lds;                   // 64 x KCHUNK f32 (32 KB)
  float* ldsZ = lds + ROWS * KCHUNK;   // 64 x VTILE  f32 (32 KB)
  __shared__ int tokLds[ROWS];

  const int tid   = threadIdx.x;
  const int tile  = blockIdx.x;
  const int vbase = tile * VTILE;

  if (tid < ROWS) tokLds[tid] = tgt[tid * N_SZ + (step - 1)];
  __syncthreads();

  const int w    = tid >> 5;
  const int lane = tid & 31;
  const int lrow = lane & 15;
  const int lhi  = lane >> 4;
  const int mt   = w & 3;   // which 16-row M tile
  const int nh   = w >> 2;  // which 64-col N half

  const v8f vzero = {0.f,0.f,0.f,0.f,0.f,0.f,0.f,0.f};
  v8f acc[4];
  #pragma unroll
  for (int nt = 0; nt < 4; ++nt) acc[nt] = vzero;

  for (int kc = 0; kc < D_SZ; kc += KCHUNK) {
    // stage gathered A chunk (emb rows of prev tokens) into LDS via async DMA:
    // 2048 b128 transfers, exactly 8 per thread -> fully uniform, no divergence.
    #pragma unroll
    for (int g = tid; g < ROWS * (KCHUNK / 4); g += 256) {
      int r  = g >> 5;   // 32 float4 per row
      int c4 = g & 31;
      const float4* src = (const float4*)(emb + (size_t)tokLds[r] * D_SZ + kc) + c4;
      unsigned ldsOff = (unsigned)(uintptr_t)(const void*)(((float4*)ldsA) + g);
      async_copy_b128(ldsOff, src);
    }
    wait_async_zero();   // this wave's LDS writes have landed
    __syncthreads();     // all waves' writes visible

    const float* arow = ldsA + (mt * 16 + lrow) * KCHUNK + 2 * lhi;
    for (int kk = 0; kk < KCHUNK; kk += 4) {
      v2f a; a.x = arow[kk]; a.y = arow[kk + 1];
      if (kc + kk + 8 < D_SZ)
        __builtin_prefetch(Wout + (size_t)(kc + kk + 8 + 2 * lhi) * V_SZ +
                               vbase + nh * 64 + lrow, 0, 1);
      #pragma unroll
      for (int nt = 0; nt < 4; ++nt) {
        int v = vbase + (nh * 4 + nt) * 16 + lrow;
        const float* bp = Wout + (size_t)(kc + kk + 2 * lhi) * V_SZ + v;
        v2f b; b.x = bp[0]; b.y = bp[V_SZ];
        acc[nt] = __builtin_amdgcn_wmma_f32_16x16x4_f32(false, a, false, b, (short)0,
                                                        acc[nt], false, false);
      }
    }
    __syncthreads();   // done reading ldsA before next chunk overwrite
  }

  // scatter C fragments (VGPR j -> row mt*16 + j + 8*lhi) into ldsZ
  #pragma unroll
  for (int nt = 0; nt < 4; ++nt) {
    int col = (nh * 4 + nt) * 16 + lrow;
    #pragma unroll
    for (int j = 0; j < 8; ++j)
      ldsZ[(mt * 16 + j + 8 * lhi) * VTILE + col] = acc[nt][j];
  }
  __syncthreads();

  if (tid < ROWS) {
    const int r = tid;
    const float* cl = CL + (size_t)(r >> 2) * V_SZ + vbase;
    float m = NEG_INF;
    for (int c = 0; c < VTILE; ++c)
      m = fmaxf(m, ldsZ[r * VTILE + c] + cl[c]);
    float s = 0.0f;
    float tv[8]; int ti[8];
    #pragma unroll
    for (int q = 0; q < 8; ++q) { tv[q] = NEG_INF; ti[q] = -1; }
    for (int c = 0; c < VTILE; ++c) {
      float z = ldsZ[r * VTILE + c] + cl[c];
      s += __expf(z - m);
      ins8(z, vbase + c, tv, ti);
    }
    pmax[tile * ROWS + r] = m;
    psum[tile * ROWS + r] = s;
    #pragma unroll
    for (int q = 0; q < 8; ++q) {
      ptopv[(tile * ROWS + r) * 8 + q] = tv[q];
      ptopi[(tile * ROWS + r) * 8 + q] = ti[q];
    }
  }
}

// ------- per-step: merge tile partials, evaluate not_done, beam update -------
__global__ void merge_update_kernel(const float* __restrict__ pmax,
                                    const float* __restrict__ psum,
                                    const float* __restrict__ ptopv,
                                    const int* __restrict__ ptopi,
                                    int* __restrict__ tgt, int* __restrict__ out,
                                    float* __restrict__ logp, float* __restrict__ score,
                                    int* __restrict__ flag, int* __restrict__ cont,
                                    int step) {
  __shared__ float rm[ROWS], rlse[ROWS], rtv[ROWS][8];
  __shared__ int rti[ROWS][8];
  __shared__ int cFlag;
  const int tid = threadIdx.x;

  if (tid < ROWS) {
    const int r = tid;
    float m = NEG_INF;
    for (int t = 0; t < T_TILES; ++t) m = fmaxf(m, pmax[t * ROWS + r]);
    float s = 0.0f;
    for (int t = 0; t < T_TILES; ++t)
      s += psum[t * ROWS + r] * __expf(pmax[t * ROWS + r] - m);
    float tv[8]; int ti[8];
    #pragma unroll
    for (int q = 0; q < 8; ++q) { tv[q] = NEG_INF; ti[q] = -1; }
    for (int t = 0; t < T_TILES; ++t) {
      #pragma unroll
      for (int q = 0; q < 8; ++q)
        ins8(ptopv[(t * ROWS + r) * 8 + q], ptopi[(t * ROWS + r) * 8 + q], tv, ti);
    }
    rm[r] = m; rlse[r] = __logf(s);
    #pragma unroll
    for (int q = 0; q < 8; ++q) { rtv[r][q] = tv[q]; rti[r][q] = ti[q]; }
  }
  __syncthreads();

  if (tid == 0) {  // not_done() on PRE-update state, global over batch
    const float penN = powf((5.0f + (float)N_SZ) / 6.0f, 0.6f);
    bool done = true;
    for (int b = 0; b < B_SZ; ++b) {
      float mn = 3.0e38f; bool fs = false;
      for (int k = 0; k < K_BEAM; ++k) {
        float p = flag[b * K_BEAM + k] ? score[b * K_BEAM + k] : 0.0f;
        mn = fminf(mn, p);
        fs = fs || (flag[b * K_BEAM + k] != 0);
      }
      float oldv = mn + (fs ? 0.0f : BIG_NEG);
      float newv = logp[b * K_BEAM] / penN;
      done = done && (oldv > newv);
    }
    int c = (*cont != 0) && !done;
    *cont = c;
    cFlag = c;
  }
  __syncthreads();

  if (tid < B_SZ) {
    const int b = tid;
    float yv[8]; int pk[8];
    #pragma unroll
    for (int q = 0; q < 8; ++q) { yv[q] = NEG_INF; pk[q] = 0; }
    for (int k = 0; k < K_BEAM; ++k) {
      int r = b * K_BEAM + k;
      float base = logp[r] - rm[r] - rlse[r];
      #pragma unroll
      for (int q = 0; q < 8; ++q)
        ins8(base + rtv[r][q], (k << 15) | (rti[r][q] & 32767), yv, pk);
    }
    const float pen = powf((5.0f + (float)(step + 1)) / 6.0f, 0.6f);
    float lpa[8], sc[8]; int bm[8], tk[8], ie[8];
    #pragma unroll
    for (int q = 0; q < 8; ++q) {
      bm[q] = pk[q] >> 15;
      tk[q] = pk[q] & 32767;
      ie[q] = (tk[q] == END_TOK) ? 1 : 0;
      lpa[q] = yv[q] + (ie[q] ? BIG_NEG : 0.0f);
      sc[q]  = yv[q] / pen + (ie[q] ? 0.0f : BIG_NEG);
    }
    int selA[4], selF[4];
    top4sel(lpa, selA);
    top4sel(sc, selF);

    int ntgt[K_BEAM][N_SZ], nout[K_BEAM][N_SZ];
    float nlogp[K_BEAM], nscore[K_BEAM];
    int nflag[K_BEAM];
    for (int j = 0; j < K_BEAM; ++j) {
      int qa = selA[j], qf = selF[j];
      int ra = b * K_BEAM + bm[qa];
      int rf = b * K_BEAM + bm[qf];
      for (int n = 0; n < N_SZ; ++n) {
        ntgt[j][n] = tgt[ra * N_SZ + n];
        nout[j][n] = tgt[rf * N_SZ + n];
      }
      ntgt[j][step] = tk[qa];
      nout[j][step] = tk[qf];
      nlogp[j]  = lpa[qa];
      nscore[j] = sc[qf];
      nflag[j]  = ie[qf];
    }
    if (cFlag) {
      for (int j = 0; j < K_BEAM; ++j) {
        int r = b * K_BEAM + j;
        for (int n = 0; n < N_SZ; ++n) {
          tgt[r * N_SZ + n] = ntgt[j][n];
          out[r * N_SZ + n] = nout[j][n];
        }
        logp[r]  = nlogp[j];
        score[r] = nscore[j];
        flag[r]  = nflag[j];
      }
    }
  }
}

// ---------------- final where(fin, out/score, tgt/logp) -> d_out ----------------
__global__ void finalize_kernel(const int* __restrict__ tgt, const int* __restrict__ out,
                                const float* __restrict__ logp, const float* __restrict__ score,
                                const int* __restrict__ flag, float* __restrict__ dout) {
  const int tid = threadIdx.x;
  for (int g = tid; g < ROWS * N_SZ; g += blockDim.x) {
    int b = (g / N_SZ) >> 2;
    bool fin = flag[b*4] || flag[b*4+1] || flag[b*4+2] || flag[b*4+3];
    dout[g] = (float)(fin ? out[g] : tgt[g]);
  }
  if (tid < ROWS) {
    int b = tid >> 2;
    bool fin = flag[b*4] || flag[b*4+1] || flag[b*4+2] || flag[b*4+3];
    dout[ROWS * N_SZ + tid] = fin ? score[tid] : logp[tid];
  }
}

extern "C" void kernel_launch(void* const* d_in, const int* in_sizes, int n_in,
                              void* d_out, int out_size, void* d_ws, size_t ws_size,
                              hipStream_t stream) {
  (void)in_sizes; (void)n_in; (void)out_size; (void)ws_size;
  const int*   x    = (const int*)d_in[0];
  const float* ctx  = (const float*)d_in[1];
  const float* emb  = (const float*)d_in[2];
  const float* Wout = (const float*)d_in[3];
  const float* Wctx = (const float*)d_in[4];

  char* ws = (char*)d_ws;
  float* CL    = (float*)(ws + OFF_CL);
  float* pmax  = (float*)(ws + OFF_PMAX);
  float* psum  = (float*)(ws + OFF_PSUM);
  float* ptopv = (float*)(ws + OFF_PTOPV);
  int*   ptopi = (int*)  (ws + OFF_PTOPI);
  int*   tgt   = (int*)  (ws + OFF_TGT);
  int*   outb  = (int*)  (ws + OFF_OUT);
  float* logp  = (float*)(ws + OFF_LOGP);
  float* score = (float*)(ws + OFF_SCORE);
  int*   flag  = (int*)  (ws + OFF_FLAG);
  int*   cont  = (int*)  (ws + OFF_CONT);

  init_state_kernel<<<1, 256, 0, stream>>>(x, tgt, outb, logp, score, flag, cont);
  ctx_logits_kernel<<<V_SZ / 64, 128, 0, stream>>>(ctx, Wctx, CL);

  const size_t ldsBytes = (size_t)(ROWS * KCHUNK + ROWS * VTILE) * sizeof(float); // 64 KB
  for (int i = 1; i < N_SZ; ++i) {
    logits_topk_kernel<<<T_TILES, 256, ldsBytes, stream>>>(
        emb, Wout, CL, tgt, i, pmax, psum, ptopv, ptopi);
    merge_update_kernel<<<1, 256, 0, stream>>>(
        pmax, psum, ptopv, ptopi, tgt, outb, logp, score, flag, cont, i);
  }

  finalize_kernel<<<1, 256, 0, stream>>>(tgt, outb, logp, score, flag, (float*)d_out);
}